// MultiHeadMemoryBank_39822936768604
// MI455X (gfx1250) — compile-verified
//
#include <hip/hip_runtime.h>
#include <math.h>

#define Bc   128
#define Hc   4
#define Nc   8192
#define Dc   64
#define TOPKc 8
#define BETAc 10.0f
#define EPSc  1e-8f

typedef __attribute__((ext_vector_type(2))) float v2f;
typedef __attribute__((ext_vector_type(8))) float v8f;

// ---------------- kernel 0: key norms (read heads 0-3, write heads 4-7) ----
__global__ void mhmb_knorm(const float* __restrict__ rk,
                           const float* __restrict__ wk,
                           float* __restrict__ kn) {
    int t = blockIdx.x * blockDim.x + threadIdx.x;
    if (t >= Bc * 8) return;
    int b = t >> 3, s = t & 7;
    const float* src = (s < 4) ? (rk + (((size_t)b * 4 + s) << 6))
                               : (wk + (((size_t)b * 4 + (s - 4)) << 6));
    float ss = 0.f;
#pragma unroll
    for (int d = 0; d < 64; ++d) ss += src[d] * src[d];
    kn[t] = fmaxf(sqrtf(ss), EPSc);
}

// ---------------- kernel 1: cosine sim via V_WMMA_F32_16X16X4_F32 ----------
// grid (N/128, B), block 256 (8 waves, each wave -> 16 memory rows)
__global__ __launch_bounds__(256) void mhmb_sim(
    const float* __restrict__ mem, const float* __restrict__ rk,
    const float* __restrict__ wk,  const float* __restrict__ kn,
    float* __restrict__ rw, float* __restrict__ ww) {
    __shared__ __align__(16) float s_keys[16 * 64];      // 4 KB
    __shared__ __align__(16) float s_tile[8][16 * 64];   // 32 KB
    const int b = blockIdx.y;
    const int t = threadIdx.x;

    // stage keys: rows 0-3 = read heads, 4-7 = write heads, 8-15 = zero
    for (int i = t; i < 16 * 64; i += 256) {
        int r = i >> 6, d = i & 63;
        float v = 0.f;
        if (r < 4)      v = rk[(((size_t)b * 4 + r) << 6) + d];
        else if (r < 8) v = wk[(((size_t)b * 4 + (r - 4)) << 6) + d];
        s_keys[i] = v;
    }

    const int wave = t >> 5, lane = t & 31;
    const int n0 = blockIdx.x * 128 + wave * 16;

    // stage 16 contiguous memory rows (16*64 floats) as 256 float4
    const float4* gsrc = (const float4*)(mem + ((size_t)b * Nc + n0) * Dc);
    float4* ldst = (float4*)&s_tile[wave][0];
    for (int i = lane; i < 256; i += 32) ldst[i] = gsrc[i];
    __syncthreads();

    // per-row memory norm: lane l (and l+16) owns row n0 + (l&15)
    float mn;
    {
        const float* row = &s_tile[wave][(lane & 15) << 6];
        float ss = 0.f;
#pragma unroll
        for (int d = 0; d < 64; ++d) ss += row[d] * row[d];
        mn = fmaxf(sqrtf(ss), EPSc);
    }

    // WMMA accumulate over K = 64 in 16 steps of 4.
    // Fragment element for lane l, step s, reg j:
    //   A: (M = l&15, K = 4s + 2*(l>>4) + j)  -> float2 from s_keys
    //   B: (N = l&15, K = 4s + 2*(l>>4) + j)  -> float2 from s_tile
    v8f acc = {0.f, 0.f, 0.f, 0.f, 0.f, 0.f, 0.f, 0.f};
    const int baseoff = ((lane & 15) << 6) + ((lane >> 4) << 1);
#pragma unroll
    for (int s = 0; s < 16; ++s) {
        v2f a  = *(const v2f*)&s_keys[baseoff + 4 * s];
        v2f bb = *(const v2f*)&s_tile[wave][baseoff + 4 * s];
        acc = __builtin_amdgcn_wmma_f32_16x16x4_f32(
            false, a, false, bb, (short)0, acc, false, false);
    }

    // C layout: lanes 0-15 hold M = reg (0..7), N = lane  -> exactly our 8 heads
    if (lane < 16) {
        const int n = n0 + lane;
#pragma unroll
        for (int r = 0; r < 8; ++r) {
            float k_n = kn[b * 8 + r];
            float v = acc[r] / (k_n * mn + EPSc) * BETAc;
            if (r < 4) rw[((size_t)b * 4 + r) * Nc + n] = v;
            else       ww[((size_t)b * 4 + (r - 4)) * Nc + n] = v;
        }
    }
}

// ---------------- kernel 2: top-8 + sparse softmax, rewrite row ------------
// grid (8, B), block 256; row staged/zeroed via b128
__global__ __launch_bounds__(256) void mhmb_topk(
    float* __restrict__ rw, float* __restrict__ ww,
    int* __restrict__ ti, float* __restrict__ tw) {
    __shared__ __align__(16) float s_row[Nc];   // 32 KB
    __shared__ float s_rv[256];
    __shared__ int   s_ri[256];
    __shared__ float s_topv[TOPKc];
    __shared__ int   s_topi[TOPKc];
    __shared__ float s_w[TOPKc];
    const int b = blockIdx.y, s = blockIdx.x, t = threadIdx.x;
    float* row = (s < 4) ? (rw + ((size_t)b * 4 + s) * Nc)
                         : (ww + ((size_t)b * 4 + (s - 4)) * Nc);
    const float4* row4 = (const float4*)row;
    float4* s_row4 = (float4*)s_row;
    for (int i = t; i < Nc / 4; i += 256) s_row4[i] = row4[i];
    __syncthreads();

    for (int k = 0; k < TOPKc; ++k) {
        float bv = -__builtin_inff(); int bi = Nc;
        for (int j = 0; j < Nc / 256; ++j) {
            int idx = t + j * 256;           // increasing -> first-max kept
            float v = s_row[idx];
            if (v > bv) { bv = v; bi = idx; }
        }
        s_rv[t] = bv; s_ri[t] = bi;
        __syncthreads();
        for (int off = 128; off > 0; off >>= 1) {
            if (t < off) {
                float v2 = s_rv[t + off]; int i2 = s_ri[t + off];
                if (v2 > s_rv[t] || (v2 == s_rv[t] && i2 < s_ri[t])) {
                    s_rv[t] = v2; s_ri[t] = i2;
                }
            }
            __syncthreads();
        }
        if (t == 0) {
            s_topv[k] = s_rv[0]; s_topi[k] = s_ri[0];
            s_row[s_ri[0]] = -__builtin_inff();
        }
        __syncthreads();
    }

    if (t == 0) {        // s_topv[0] is the global max
        float m = s_topv[0], sum = 0.f, e[TOPKc];
#pragma unroll
        for (int k = 0; k < TOPKc; ++k) { e[k] = expf(s_topv[k] - m); sum += e[k]; }
#pragma unroll
        for (int k = 0; k < TOPKc; ++k) {
            float w = e[k] / sum;
            s_w[k] = w;
            int g = (b * 8 + s) * 8 + k;
            tw[g] = w; ti[g] = s_topi[k];
        }
    }
    __syncthreads();
    float4 z = {0.f, 0.f, 0.f, 0.f};
    float4* out4 = (float4*)row;
    for (int i = t; i < Nc / 4; i += 256) out4[i] = z;  // masked softmax -> 0
    __syncthreads();
    if (t < TOPKc) row[s_topi[t]] = s_w[t];
}

// ---------------- kernel 3: read = mean_h sum_k w * mem[idx] ---------------
__global__ void mhmb_read(const float* __restrict__ mem,
                          const int* __restrict__ ti,
                          const float* __restrict__ tw,
                          float* __restrict__ rc) {
    int b = blockIdx.x, d = threadIdx.x;
    float acc = 0.f;
    for (int h = 0; h < 4; ++h)
#pragma unroll
        for (int k = 0; k < TOPKc; ++k) {
            int g = (b * 8 + h) * 8 + k;
            acc += tw[g] * mem[((size_t)b * Nc + ti[g]) * Dc + d];
        }
    rc[b * Dc + d] = acc * 0.25f;
}

// ---------------- kernel 4: new_mem = 4*m - m*E + A (b128 streaming) -------
// grid B*N/16 blocks of 256; one float4 per thread, 16 rows per block
__global__ __launch_bounds__(256) void mhmb_update(
    const float* __restrict__ mem, const float* __restrict__ er,
    const float* __restrict__ wv,  const int* __restrict__ ti,
    const float* __restrict__ tw,  float* __restrict__ nm) {
    __shared__ int   s_i[32];
    __shared__ float s_wt[32];
    const int bid = blockIdx.x;
    const int b = bid / (Nc / 16);
    const int n_base = (bid % (Nc / 16)) * 16;
    const int t = threadIdx.x;
    if (t < 32) {                 // write-head (heads 4..7) topk pairs
        int g = (b * 8 + 4 + (t >> 3)) * 8 + (t & 7);
        s_i[t] = ti[g]; s_wt[t] = tw[g];
    }
    __syncthreads();
    const int n = n_base + (t >> 4);   // row; uniform across each 16-lane group
    const int q = t & 15;              // float4 slot within the 64-float row
    const size_t off4 = ((size_t)b * Nc + n) * (Dc / 4) + q;
    const float4 m = ((const float4*)mem)[off4];
    float4 E = {0.f, 0.f, 0.f, 0.f}, A = {0.f, 0.f, 0.f, 0.f};
    for (int h = 0; h < 4; ++h) {
#pragma unroll
        for (int k = 0; k < 8; ++k) {
            int j = h * 8 + k;
            if (s_i[j] == n) {
                float w = s_wt[j];
                float4 e4 = ((const float4*)er)[((size_t)b * 4 + h) * (Dc / 4) + q];
                float4 v4 = ((const float4*)wv)[((size_t)b * 4 + h) * (Dc / 4) + q];
                E.x += w * e4.x; E.y += w * e4.y; E.z += w * e4.z; E.w += w * e4.w;
                A.x += w * v4.x; A.y += w * v4.y; A.z += w * v4.z; A.w += w * v4.w;
            }
        }
    }
    float4 o;
    o.x = 4.0f * m.x - m.x * E.x + A.x;
    o.y = 4.0f * m.y - m.y * E.y + A.y;
    o.z = 4.0f * m.z - m.z * E.z + A.z;
    o.w = 4.0f * m.w - m.w * E.w + A.w;
    ((float4*)nm)[off4] = o;
}

extern "C" void kernel_launch(void* const* d_in, const int* in_sizes, int n_in,
                              void* d_out, int out_size, void* d_ws, size_t ws_size,
                              hipStream_t stream) {
    const float* mem = (const float*)d_in[0];
    const float* rk  = (const float*)d_in[1];
    const float* wk  = (const float*)d_in[2];
    const float* wv  = (const float*)d_in[3];
    const float* er  = (const float*)d_in[4];

    float* out = (float*)d_out;
    float* rc = out;                                    // (B,D)
    float* rw = out + (size_t)Bc * Dc;                  // (B,H,N)
    float* nm = rw + (size_t)Bc * Hc * Nc;              // (B,N,D)
    float* ww = nm + (size_t)Bc * Nc * Dc;              // (B,H,N)

    float* kn = (float*)d_ws;                           // 1024 floats
    int*   ti = (int*)((char*)d_ws + 4096);             // 8192 ints
    float* tw = (float*)((char*)d_ws + 4096 + 32768);   // 8192 floats

    mhmb_knorm<<<4, 256, 0, stream>>>(rk, wk, kn);
    mhmb_sim<<<dim3(Nc / 128, Bc), 256, 0, stream>>>(mem, rk, wk, kn, rw, ww);
    mhmb_topk<<<dim3(8, Bc), 256, 0, stream>>>(rw, ww, ti, tw);
    mhmb_read<<<Bc, 64, 0, stream>>>(mem, ti, tw, rc);
    mhmb_update<<<Bc * (Nc / 16), 256, 0, stream>>>(mem, er, wv, ti, tw, nm);
}